// MemoryCell_52793738002525
// MI455X (gfx1250) — compile-verified
//
#include <hip/hip_runtime.h>
#include <cstdint>
#include <cstddef>

// ---------------- problem / tiling constants ----------------
constexpr int kB        = 4096;              // batch rows
constexpr int kT        = 4096;              // timesteps
constexpr int kTT       = 32;                // timesteps per LDS tile
constexpr int kNTiles   = kT / kTT;          // 128
constexpr int kWavesPB  = 2;
constexpr int kBlock    = kWavesPB * 32;     // 64 threads
constexpr int kRowsPB   = kBlock;            // one row per thread
constexpr int kGrid     = kB / kRowsPB;      // 64 blocks
constexpr int kRowBytes = kTT * 2 * 4;       // 256 B of (a,b) pairs per row per tile
constexpr int kPitch    = kRowBytes + 16;    // 272 B LDS row pitch (16B aligned)
constexpr int kInBuf    = 32 * kPitch;       // 8704 B per wave input buffer
constexpr int kOutBuf   = 32 * kPitch;       // 8704 B per wave output staging
constexpr int kWaveLds  = 2 * kInBuf + kOutBuf;   // 26112 B / wave
constexpr int kBlockLds = kWavesPB * kWaveLds;    // 52224 B / block (< 64KB)

typedef unsigned int u32x4 __attribute__((ext_vector_type(4)));
typedef int          i32x4 __attribute__((ext_vector_type(4)));
typedef int          i32x8 __attribute__((ext_vector_type(8)));

// ---------------- synapse params, pre-folded for exp2 ----------------
struct SynP { float g, c0, c1, pot; };

__device__ __forceinline__ SynP mk_syn(const float* __restrict__ p, int i) {
    const float L2E = 1.4426950408889634f;
    SynP s;
    float g = p[i], mean = p[i + 1], sd = p[i + 2], pot = p[i + 3];
    s.g  = g;
    s.c1 = -sd * L2E;        // sigmoid(sd*(pre-mean)) = 1/(1+exp2(c1*pre + c0))
    s.c0 =  sd * mean * L2E;
    s.pot = pot;
    return s;
}

__device__ __forceinline__ float syn_eval(const SynP s, float pre, float post) {
    float e  = __builtin_amdgcn_exp2f(__builtin_fmaf(s.c1, pre, s.c0)); // v_exp_f32
    float sg = __builtin_amdgcn_rcpf(1.0f + e);                         // v_rcp_f32
    return s.g * sg * (s.pot - post);
}

// ---------------- one TDM tile load: 32 rows x 256B, strided, pad->pitch 272 ----------------
__device__ __forceinline__ void tdm_load_tile(uint32_t lds_addr, uint64_t gaddr) {
    u32x4 g0;
    g0.x = 1u;                                      // count=1, user descriptor, no gather
    g0.y = lds_addr;                                // LDS byte address of tile start
    g0.z = (uint32_t)gaddr;                         // global_addr[31:0]
    g0.w = ((uint32_t)(gaddr >> 32) & 0x01FFFFFFu)  // global_addr[56:32]
         | (2u << 30);                              // type = 2 ("image")
    i32x8 g1;
    g1.s0 = (int)((2u << 16)    // data_size = 4 bytes
                | (1u << 20)    // pad_enable: insert LDS padding on load
                | (5u << 22)    // pad_interval: every 64 DWORDs (=256B row)
                | (3u << 25));  // pad_amount: 4 DWORDs (16B) -> pitch 272
    g1.s1 = (int)((uint32_t)(kTT * 2) << 16);   // tensor_dim0 = 64 elems (bits 79:48 lo)
    g1.s2 = (int)(32u << 16);                   // tensor_dim1 = 32 rows  (bits 111:80 lo)
    g1.s3 = (int)((uint32_t)(kTT * 2) << 16);   // tile_dim0  = 64 elems  (bits 127:112)
    g1.s4 = 32;                                 // tile_dim1  = 32 rows   (bits 143:128)
    g1.s5 = kT * 2;                             // tensor_dim0_stride = 8192 elems
    g1.s6 = 0;
    g1.s7 = 0;
    i32x4 gz = {0, 0, 0, 0};
#if defined(__clang_major__) && (__clang_major__ >= 23)
    i32x8 gz8 = {0, 0, 0, 0, 0, 0, 0, 0};
    __builtin_amdgcn_tensor_load_to_lds(g0, g1, gz, gz, gz8, 0);
#else
    __builtin_amdgcn_tensor_load_to_lds(g0, g1, gz, gz, 0);
#endif
}

// ---------------- kernel: one batch row per lane, TDM double-buffered over time ----------------
__global__ __launch_bounds__(kBlock) void
memcell_tdm_kernel(const float* __restrict__ inp,
                   const float* __restrict__ prm,
                   float* __restrict__ outp) {
    extern __shared__ __align__(16) char smem[];

    const int tid  = (int)threadIdx.x;
    const int lane = tid & 31;
    const uint32_t wv = __builtin_amdgcn_readfirstlane((uint32_t)tid >> 5);

    // per-wave LDS carve-out (wave fully self-contained -> no barriers)
    char* pInA = smem + (size_t)wv * kWaveLds;
    char* pInB = pInA + kInBuf;
    char* pOut = pInA + 2 * kInBuf;
    const uint32_t ldsInA = (uint32_t)(uintptr_t)pInA;  // generic ptr low 32 = LDS byte addr
    const uint32_t ldsInB = (uint32_t)(uintptr_t)pInB;

    // first global row handled by this wave (uniform per wave -> SGPR for TDM descriptor)
    const uint32_t rowbaseW = __builtin_amdgcn_readfirstlane(
        (uint32_t)(blockIdx.x * kRowsPB) + wv * 32u);
    const uint64_t gRow0 = (uint64_t)(uintptr_t)inp
                         + (uint64_t)rowbaseW * (uint64_t)(kT * 2) * 4ull;

    // fold params (uniform scalar loads)
    const float icx = __builtin_amdgcn_rcpf(prm[0]);
    const float icy = __builtin_amdgcn_rcpf(prm[1]);
    const SynP s_ax = mk_syn(prm, 2);    // dir ax = 0
    const SynP s_by = mk_syn(prm, 6);    // dir by = 1
    const SynP s_xy = mk_syn(prm, 10);   // dir xy = 2
    const SynP s_yx = mk_syn(prm, 14);   // dir yx = 3
    const SynP s_xx = mk_syn(prm, 18);   // dir xx = 4
    const SynP s_yy = mk_syn(prm, 22);   // dir yy = 5

    float x = 0.0f, y = 1.0f;            // reference initial state

    // prefetch tile 0
    __asm__ volatile("" ::: "memory");
    tdm_load_tile(ldsInA, gRow0);

    const float2* irowA = (const float2*)(pInA + lane * kPitch);
    const float2* irowB = (const float2*)(pInB + lane * kPitch);
    float2*       orow  = (float2*)(pOut + lane * kPitch);

    const int sub = lane >> 4;           // transpose helpers: 2 rows per b128 op
    const int ch  = (lane & 15) << 4;    // 16B chunk within a row

    for (int tile = 0; tile < kNTiles; ++tile) {
        const float2* irow = (tile & 1) ? irowB : irowA;

        if (tile + 1 < kNTiles) {
            // issue next tile's DMA into the other buffer, then wait for current tile
            __asm__ volatile("" ::: "memory");
            tdm_load_tile((tile & 1) ? ldsInA : ldsInB,
                          gRow0 + (uint64_t)(tile + 1) * (uint64_t)(kTT * 2 * 4));
            __builtin_amdgcn_s_wait_tensorcnt((short)1);
        } else {
            __builtin_amdgcn_s_wait_tensorcnt((short)0);
        }
        __asm__ volatile("" ::: "memory");   // LDS now holds tile `tile`

        // ---- recurrence over this tile (all state in registers) ----
#pragma unroll 4
        for (int t = 0; t < kTT; ++t) {
            float2 ab = irow[t];
            float a = ab.x, b = ab.y;
            float cx = syn_eval(s_ax, a, x) + syn_eval(s_yx, y, x) + syn_eval(s_xx, x, x);
            float cy = syn_eval(s_by, b, y) + syn_eval(s_xy, x, y) + syn_eval(s_yy, y, y);
            x = __builtin_fmaf(cx, icx, x);
            y = __builtin_fmaf(cy, icy, y);
            orow[t] = make_float2(x, y);
        }

        // ---- transpose out-tile and store coalesced (256B contiguous per row) ----
        const long t0 = (long)tile * kTT;
#pragma unroll
        for (int r = 0; r < 32; r += 2) {
            const int row = r + sub;
            float4 v = *(const float4*)(pOut + row * kPitch + ch);
            char* dst = (char*)(outp + ((size_t)(rowbaseW + row) * (size_t)(kT * 2)
                                        + (size_t)t0 * 2)) + ch;
            *(float4*)dst = v;
        }
    }
}

// ---------------- host entry ----------------
extern "C" void kernel_launch(void* const* d_in, const int* in_sizes, int n_in,
                              void* d_out, int out_size, void* d_ws, size_t ws_size,
                              hipStream_t stream) {
    (void)in_sizes; (void)n_in; (void)out_size; (void)d_ws; (void)ws_size;
    const float* inp = (const float*)d_in[0];
    const float* prm = (const float*)d_in[1];
    float*       out = (float*)d_out;
    hipLaunchKernelGGL(memcell_tdm_kernel, dim3(kGrid), dim3(kBlock), kBlockLds, stream,
                       inp, prm, out);
}